// UpdateFunction_4698694222685
// MI455X (gfx1250) — compile-verified
//
#include <hip/hip_runtime.h>

#define LMAX 5
#define NTRIP 69

// d_ws layout offsets, in 4-byte words
#define OFF_TL1   0
#define OFF_TL2   96
#define OFF_TL    192
#define OFF_TRANK 288
#define OFF_LCNT  384
#define OFF_LTRIP 392        // 6 * 16
#define OFF_LK_L  488        // 36
#define OFF_LK_K  524        // 36
#define OFF_NCOLS 560        // 6
#define OFF_CGCNT 568        // 69*11 = 759
#define OFF_CGPK  1328       // 69*11*12 = 9108
#define OFF_CGCF  10440      // 9108
#define WS_WORDS  19552

typedef float v2f __attribute__((ext_vector_type(2)));
typedef float v8f __attribute__((ext_vector_type(8)));

struct P12 { const float* f[6]; const float* w[6]; };

__device__ __forceinline__ int rfl(int x) { return __builtin_amdgcn_readfirstlane(x); }
__device__ __forceinline__ float rflf(float x) {
  return __builtin_bit_cast(float, __builtin_amdgcn_readfirstlane(__builtin_bit_cast(int, x)));
}

// ---------------- CG coefficient (Racah), double precision ----------------
__device__ __forceinline__ double dfact(int n) {
  const double F[17] = {1.,1.,2.,6.,24.,120.,720.,5040.,40320.,362880.,3628800.,
                        39916800.,479001600.,6227020800.,87178291200.,
                        1307674368000.,20922789888000.};
  return F[n];
}

__device__ float cg_coeff_dev(int l1,int l2,int l,int m1,int m2,int m) {
  double pref = sqrt((2.0*l+1.0)*dfact(l+l1-l2)*dfact(l-l1+l2)*dfact(l1+l2-l)
                     / dfact(l1+l2+l+1));
  pref *= sqrt(dfact(l+m)*dfact(l-m)*dfact(l1-m1)*dfact(l1+m1)
               *dfact(l2-m2)*dfact(l2+m2));
  int kmin = max(0, max(l2-l-m1, l1+m2-l));
  int kmax = min(l1+l2-l, min(l1-m1, l2+m2));
  double s = 0.0;
  for (int kk = kmin; kk <= kmax; ++kk) {
    double d = dfact(kk)*dfact(l1+l2-l-kk)*dfact(l1-m1-kk)*dfact(l2+m2-kk)
             * dfact(l-l2+m1+kk)*dfact(l-l1-m2+kk);
    s += ((kk & 1) ? -1.0 : 1.0) / d;
  }
  return (float)(pref * s);
}

// ---------------- Init kernel: build triple + CG tables in d_ws ----------------
__global__ void init_tables(int* __restrict__ wsI, float* __restrict__ wsF) {
  const int tid = threadIdx.x;
  if (tid == 0) {
    int cnt = 0;
    int rcnt[LMAX+1];
    for (int i = 0; i <= LMAX; ++i) rcnt[i] = 0;
    for (int l1 = 0; l1 <= LMAX; ++l1)
      for (int l2 = 0; l2 <= l1; ++l2)
        for (int l = l1 - l2; l <= min(l1 + l2, LMAX); ++l) {
          wsI[OFF_TL1 + cnt] = l1;
          wsI[OFF_TL2 + cnt] = l2;
          wsI[OFF_TL  + cnt] = l;
          wsI[OFF_TRANK + cnt] = rcnt[l];
          wsI[OFF_LTRIP + l*16 + rcnt[l]] = cnt;
          rcnt[l]++; cnt++;
        }
    for (int l = 0; l <= LMAX; ++l) {
      wsI[OFF_LCNT + l] = rcnt[l];
      wsI[OFF_NCOLS + l] = rcnt[l] * 256;
    }
    int idx = 0;
    for (int l = 0; l <= LMAX; ++l)
      for (int k = 0; k < 2*l+1; ++k) { wsI[OFF_LK_L+idx] = l; wsI[OFF_LK_K+idx] = k; idx++; }
  }
  // Each thread < NTRIP independently re-enumerates to find its own triple.
  if (tid < NTRIP) {
    int cnt = 0, myl1 = 0, myl2 = 0, myl = 0;
    for (int l1 = 0; l1 <= LMAX; ++l1)
      for (int l2 = 0; l2 <= l1; ++l2)
        for (int l = l1 - l2; l <= min(l1 + l2, LMAX); ++l) {
          if (cnt == tid) { myl1 = l1; myl2 = l2; myl = l; }
          cnt++;
        }
    for (int k = 0; k < 2*myl + 1; ++k) {
      const int m = k - myl;
      const int base = (tid*11 + k)*12;
      int n = 0;
      for (int m1 = -myl1; m1 <= myl1; ++m1) {
        const int m2 = m - m1;
        if (m2 < -myl2 || m2 > myl2) continue;
        wsI[OFF_CGPK + base + n] = (m1 + myl1) | ((m2 + myl2) << 8);
        wsF[OFF_CGCF + base + n] = cg_coeff_dev(myl1, myl2, myl, m1, m2, m);
        n++;
      }
      wsI[OFF_CGCNT + tid*11 + k] = n;
      // zero-pad: main kernel always iterates 11 entries branch-free
      for (; n < 12; ++n) { wsI[OFF_CGPK + base + n] = 0; wsF[OFF_CGCF + base + n] = 0.f; }
    }
  }
}

// ---------------- Fused CG-transform + complex channel-mix (WMMA fp32) ----------------
// Grid: blockIdx.x in [0, 36*16): lk = x % 36 -> (l,k);  b_tile = x / 36 (16 batch each).
// Block: 128 threads = 4 waves; waves split the triples of l.
// Complex GEMM out[t,b] += W[t,c]*Frag[c,b] via V_WMMA_F32_16X16X4_F32 (4 real GEMMs).
// All table values forced into SGPRs (readfirstlane) -> every loop is wave-uniform,
// EXEC stays all-ones through the WMMAs, CG list lives in scalar registers per triple.
// jc unrolled by 2 (not 4) to stay under 256 VGPRs (avoid s_set_vgpr_msb + keep occupancy).
__global__ __launch_bounds__(128) void cg_fused(
    P12 p, const int* __restrict__ wsI, const float* __restrict__ wsF,
    float* __restrict__ out)
{
  __shared__ float red[4][32][16];

  const int lk    = blockIdx.x % 36;
  const int btile = blockIdx.x / 36;
  const int l  = rfl(wsI[OFF_LK_L + lk]);
  const int k  = rfl(wsI[OFF_LK_K + lk]);
  const int Cl = rfl(wsI[OFF_NCOLS + l]);
  const int nT = rfl(wsI[OFF_LCNT + l]);
  const int dl = 2*l + 1;

  const int wave = threadIdx.x >> 5;
  const int lane = threadIdx.x & 31;
  const int hsel = lane >> 4;     // 0: K-cols 0,1 / D rows 0-7 ; 1: K-cols 2,3 / D rows 8-15
  const int lrow = lane & 15;     // A: row t ; B/D: column b_local
  const int b    = btile * 16 + lrow;

  const float* __restrict__ W = p.w[l];

  v8f rr = {}, ii = {}, ri = {}, ir = {};

  for (int ti = wave; ti < nT; ti += 4) {
    const int trip = rfl(wsI[OFF_LTRIP + l*16 + ti]);
    const int l1   = rfl(wsI[OFF_TL1 + trip]);
    const int l2   = rfl(wsI[OFF_TL2 + trip]);
    const int colb = rfl(wsI[OFF_TRANK + trip]) * 256;
    const int d1 = 2*l1 + 1, d2 = 2*l2 + 1;
    const float* __restrict__ f1b = p.f[l1] + (b*16) * d1 * 2;
    const float* __restrict__ f2b = p.f[l2] + (b*16) * d2 * 2;
    const int cgbase = (trip*11 + k) * 12;

    // CG nonzero list -> scalar registers (zero-padded to 11, branch-free)
    int   i1s[11], i2s[11];
    float cfs[11];
#pragma unroll
    for (int e = 0; e < 11; ++e) {
      const int pk = rfl(wsI[OFF_CGPK + cgbase + e]);
      i1s[e] = pk & 255;
      i2s[e] = pk >> 8;
      cfs[e] = rflf(wsF[OFF_CGCF + cgbase + e]);
    }

    for (int i = 0; i < 16; ++i) {            // f1 channel (c / 16)
      const float* __restrict__ r1 = f1b + i * d1 * 2;
      float2 av[11];                          // f1 values, shared by 4 chunks below
#pragma unroll
      for (int e = 0; e < 11; ++e) av[e] = *(const float2*)(r1 + i1s[e]*2);

#pragma unroll 2
      for (int jc = 0; jc < 4; ++jc) {        // 4 K-chunks share this i
        const int c0 = i*16 + jc*4;
        const int cA = c0 + hsel*2;           // this lane's first K-col
        const int jA = cA & 15;               // f2 channel (<=14, jA+1 valid)

        // A-operand: weight tile, one b128 per lane (c and c+1 real/imag pairs)
        const float4 wl = *(const float4*)(W + ((size_t)lrow * Cl + colb + cA) * 2);

        // B-operand: two frag values (complex) from the scalar CG list
        float frA = 0.f, fiA = 0.f, frB = 0.f, fiB = 0.f;
        const float* __restrict__ r2  = f2b + jA * d2 * 2;
        const float* __restrict__ r2b = r2 + d2 * 2;   // channel jA+1
#pragma unroll
        for (int e = 0; e < 11; ++e) {
          const float2 a  = av[e];
          const float  cf = cfs[e];
          const float2 u  = *(const float2*)(r2  + i2s[e]*2);
          const float2 vv = *(const float2*)(r2b + i2s[e]*2);
          frA += cf * (a.x*u.x  - a.y*u.y);
          fiA += cf * (a.x*u.y  + a.y*u.x);
          frB += cf * (a.x*vv.x - a.y*vv.y);
          fiB += cf * (a.x*vv.y + a.y*vv.x);
        }
        v2f a_r = { wl.x, wl.z };
        v2f a_i = { wl.y, wl.w };
        v2f b_r = { frA, frB };
        v2f b_i = { fiA, fiB };

        // Complex GEMM as 4 real fp32 WMMAs
        rr = __builtin_amdgcn_wmma_f32_16x16x4_f32(false, a_r, false, b_r, (short)0, rr, false, false);
        ii = __builtin_amdgcn_wmma_f32_16x16x4_f32(false, a_i, false, b_i, (short)0, ii, false, false);
        ri = __builtin_amdgcn_wmma_f32_16x16x4_f32(false, a_r, false, b_i, (short)0, ri, false, false);
        ir = __builtin_amdgcn_wmma_f32_16x16x4_f32(false, a_i, false, b_r, (short)0, ir, false, false);
      }
    }
  }

  // Combine complex parts, stage per-wave partials in LDS
#pragma unroll
  for (int v = 0; v < 8; ++v) {
    red[wave][lane][v]     = rr[v] - ii[v];
    red[wave][lane][8 + v] = ri[v] + ir[v];
  }
  __syncthreads();

  // Wave 0 reduces across the 4 waves and writes out.
  // D layout: lane<16 -> (N=lane, M=vgpr v); lane>=16 -> (N=lane-16, M=v+8)
  if (threadIdx.x < 32) {
#pragma unroll
    for (int v = 0; v < 8; ++v) {
      const float re = red[0][lane][v]   + red[1][lane][v]
                     + red[2][lane][v]   + red[3][lane][v];
      const float im = red[0][lane][8+v] + red[1][lane][8+v]
                     + red[2][lane][8+v] + red[3][lane][8+v];
      const int t = v + 8 * hsel;
      float* o = out + 8192 * l * l + ((size_t)(b * 16 + t) * dl + k) * 2;
      o[0] = re;
      o[1] = im;
    }
  }
}

extern "C" void kernel_launch(void* const* d_in, const int* in_sizes, int n_in,
                              void* d_out, int out_size, void* d_ws, size_t ws_size,
                              hipStream_t stream) {
  (void)in_sizes; (void)n_in; (void)out_size; (void)ws_size;
  int*   wsI = (int*)d_ws;
  float* wsF = (float*)d_ws;

  hipLaunchKernelGGL(init_tables, dim3(1), dim3(128), 0, stream, wsI, wsF);

  P12 p;
  for (int l = 0; l < 6; ++l) {
    p.f[l] = (const float*)d_in[2*l];       // inputs interleave: f0,w0,f1,w1,...
    p.w[l] = (const float*)d_in[2*l + 1];
  }
  // 36 (l,k) pairs x 16 batch tiles
  hipLaunchKernelGGL(cg_fused, dim3(36 * 16), dim3(128), 0, stream,
                     p, (const int*)d_ws, (const float*)d_ws, (float*)d_out);
}